// MultiHeadAttention_62371515073076
// MI455X (gfx1250) — compile-verified
//
#include <hip/hip_runtime.h>
#include <math.h>

// ---------------------------------------------------------------------------
// MultiHeadAttention (quirky softmax-over-query variant) for gfx1250 / MI455X
// f32 WMMA (V_WMMA_F32_16X16X4_F32) everywhere; two-pass online column-
// softmax so the 256MB S matrix is never materialized; 32x32 register
// blocking per wave (4 WMMA accumulators) to halve vmem-per-wmma.
// ---------------------------------------------------------------------------

typedef float v2f __attribute__((ext_vector_type(2)));
typedef float v8f __attribute__((ext_vector_type(8)));

#define WMMA_F32(a, b, c) \
  __builtin_amdgcn_wmma_f32_16x16x4_f32(false, (a), false, (b), (short)0, (c), false, false)

constexpr int BATCH = 8;
constexpr int SEQ   = 1024;
constexpr int CIN   = 256;
constexpr int HEADS = 8;
constexpr int DIM   = 64;
constexpr int HD    = HEADS * DIM;   // 512
constexpr int COUT  = 256;
constexpr float SCALE = 0.125f;      // 1/sqrt(64)

// ---------------------------------------------------------------------------
// Kernel 1: fused QKV projection, 32x32 tile per wave.
//   O[b] (1024x512) = X[b] (1024x256) * W_sel (256x512), sel in {q,k,v}
// Per K-step (K=4): 2 A-frags (float2 each) + 2 B-frags -> 4 WMMAs.
// ---------------------------------------------------------------------------
__global__ __launch_bounds__(32) void qkv_gemm(
    const float* __restrict__ X,
    const float* __restrict__ Wq, const float* __restrict__ Wk,
    const float* __restrict__ Wv,
    float* __restrict__ qws, float* __restrict__ kws, float* __restrict__ vws)
{
  const int lane = threadIdx.x;
  const int half = lane >> 4, l16 = lane & 15;
  const int n0 = blockIdx.x * 32;
  const int i0 = blockIdx.y * 32;
  const int b  = blockIdx.z / 3;
  const int sel = blockIdx.z % 3;

  const float* W = (sel == 0) ? Wq : (sel == 1) ? Wk : Wv;
  float*       O = (sel == 0) ? qws : (sel == 1) ? kws : vws;

  const float* xrow0 = X + (size_t)(b * SEQ + i0 + l16) * CIN;
  const float* xrow1 = xrow0 + (size_t)16 * CIN;

  v8f acc00 = {}, acc01 = {}, acc10 = {}, acc11 = {};
#pragma unroll 8
  for (int kb = 0; kb < CIN; kb += 4) {
    const int kk = kb + half * 2;
    v2f a0 = *(const v2f*)(xrow0 + kk);
    v2f a1 = *(const v2f*)(xrow1 + kk);
    v2f b0, b1;
    b0.x = W[(size_t)kk * HD + n0 + l16];
    b0.y = W[(size_t)(kk + 1) * HD + n0 + l16];
    b1.x = W[(size_t)kk * HD + n0 + 16 + l16];
    b1.y = W[(size_t)(kk + 1) * HD + n0 + 16 + l16];
    acc00 = WMMA_F32(a0, b0, acc00);
    acc01 = WMMA_F32(a0, b1, acc01);
    acc10 = WMMA_F32(a1, b0, acc10);
    acc11 = WMMA_F32(a1, b1, acc11);
  }
#pragma unroll
  for (int r = 0; r < 8; ++r) {
    const size_t row0 = (size_t)(b * SEQ + i0 + r + half * 8) * HD;
    const size_t row1 = (size_t)(b * SEQ + i0 + 16 + r + half * 8) * HD;
    O[row0 + n0 + l16]      = acc00[r];
    O[row0 + n0 + 16 + l16] = acc01[r];
    O[row1 + n0 + l16]      = acc10[r];
    O[row1 + n0 + 16 + l16] = acc11[r];
  }
}

// ---------------------------------------------------------------------------
// Kernel 2: pass 1 of column-softmax (softmax over query axis i per key j).
// One wave owns (b, h, 32 columns j); both B-fragment sets hoisted (2x32
// VGPRs); streamed Q A-fragment shared by the two column tiles -> 16 loads
// feed 32 WMMAs per i-tile. Online max/sum per column.
// ---------------------------------------------------------------------------
__global__ __launch_bounds__(32) void col_stats(
    const float* __restrict__ qws, const float* __restrict__ kws,
    float* __restrict__ mcol, float* __restrict__ lcol)
{
  const int lane = threadIdx.x, half = lane >> 4, l16 = lane & 15;
  const int j0 = blockIdx.x * 32;
  const int h  = blockIdx.y;
  const int b  = blockIdx.z;
  const int jA = j0 + l16;
  const int jB = j0 + 16 + l16;

  const float* kpA = kws + (size_t)(b * SEQ + jA) * HD + h * DIM;
  const float* kpB = kws + (size_t)(b * SEQ + jB) * HD + h * DIM;
  v2f bfA[16], bfB[16];
#pragma unroll
  for (int t = 0; t < 16; ++t) {
    bfA[t] = *(const v2f*)(kpA + t * 4 + half * 2);
    bfB[t] = *(const v2f*)(kpB + t * 4 + half * 2);
  }

  float m0 = -__builtin_inff(), l0 = 0.0f;
  float m1 = -__builtin_inff(), l1 = 0.0f;
  const float* qbase = qws + (size_t)b * SEQ * HD + h * DIM;

  for (int i0 = 0; i0 < SEQ; i0 += 16) {
    const float* qrow = qbase + (size_t)(i0 + l16) * HD;
    v8f accA = {}, accB = {};
#pragma unroll
    for (int t = 0; t < 16; ++t) {
      v2f a = *(const v2f*)(qrow + t * 4 + half * 2);
      accA = WMMA_F32(a, bfA[t], accA);
      accB = WMMA_F32(a, bfB[t], accB);
    }
    // lane L holds 8 rows of column j; lane L^16 holds the other 8 rows.
    float sA[8], sB[8];
    float tmA = -__builtin_inff(), tmB = -__builtin_inff();
#pragma unroll
    for (int r = 0; r < 8; ++r) {
      sA[r] = accA[r] * SCALE; tmA = fmaxf(tmA, sA[r]);
      sB[r] = accB[r] * SCALE; tmB = fmaxf(tmB, sB[r]);
    }
    tmA = fmaxf(tmA, __shfl_xor(tmA, 16, 32));
    tmB = fmaxf(tmB, __shfl_xor(tmB, 16, 32));
    const float mn0 = fmaxf(m0, tmA);
    const float mn1 = fmaxf(m1, tmB);
    float psA = 0.0f, psB = 0.0f;
#pragma unroll
    for (int r = 0; r < 8; ++r) {
      psA += __expf(sA[r] - mn0);
      psB += __expf(sB[r] - mn1);
    }
    psA += __shfl_xor(psA, 16, 32);
    psB += __shfl_xor(psB, 16, 32);
    l0 = l0 * __expf(m0 - mn0) + psA;   // exp(-inf)=0 on first tile
    l1 = l1 * __expf(m1 - mn1) + psB;
    m0 = mn0;
    m1 = mn1;
  }

  if (half == 0) {
    const size_t base = (size_t)(b * HEADS + h) * SEQ;
    mcol[base + jA] = m0;  lcol[base + jA] = l0;
    mcol[base + jB] = m1;  lcol[base + jB] = l1;
  }
}

// ---------------------------------------------------------------------------
// Kernel 3: pass 2 — w[b,i,h] = sum_j exp(S[i,j]-m[j]) / l[j].
// One wave owns (b, h, 32 rows i); both A-fragment sets hoisted; streamed
// K-fragment + column stats shared by the two row tiles.
// ---------------------------------------------------------------------------
__global__ __launch_bounds__(32) void row_wsum(
    const float* __restrict__ qws, const float* __restrict__ kws,
    const float* __restrict__ mcol, const float* __restrict__ lcol,
    float* __restrict__ wsum)
{
  const int lane = threadIdx.x, half = lane >> 4, l16 = lane & 15;
  const int i0 = blockIdx.x * 32;
  const int h  = blockIdx.y;
  const int b  = blockIdx.z;

  const float* qrowA = qws + (size_t)(b * SEQ + i0 + l16) * HD + h * DIM;
  const float* qrowB = qrowA + (size_t)16 * HD;
  v2f afA[16], afB[16];
#pragma unroll
  for (int t = 0; t < 16; ++t) {
    afA[t] = *(const v2f*)(qrowA + t * 4 + half * 2);
    afB[t] = *(const v2f*)(qrowB + t * 4 + half * 2);
  }

  float waccA[8], waccB[8];
#pragma unroll
  for (int r = 0; r < 8; ++r) { waccA[r] = 0.0f; waccB[r] = 0.0f; }

  const float* kbase = kws + (size_t)b * SEQ * HD + h * DIM;
  const float* mrow  = mcol + (size_t)(b * HEADS + h) * SEQ;
  const float* lrow  = lcol + (size_t)(b * HEADS + h) * SEQ;

  for (int j0 = 0; j0 < SEQ; j0 += 16) {
    const int j = j0 + l16;
    const float mj   = mrow[j];
    const float invl = 1.0f / lrow[j];
    const float* krow = kbase + (size_t)j * HD;
    v8f accA = {}, accB = {};
#pragma unroll
    for (int t = 0; t < 16; ++t) {
      v2f bb = *(const v2f*)(krow + t * 4 + half * 2);
      accA = WMMA_F32(afA[t], bb, accA);
      accB = WMMA_F32(afB[t], bb, accB);
    }
#pragma unroll
    for (int r = 0; r < 8; ++r) {
      waccA[r] += __expf(accA[r] * SCALE - mj) * invl;
      waccB[r] += __expf(accB[r] * SCALE - mj) * invl;
    }
  }

  // Row r lives in the 16 lanes of one half -> butterfly reduce within half.
#pragma unroll
  for (int r = 0; r < 8; ++r) {
#pragma unroll
    for (int mask = 1; mask < 16; mask <<= 1) {
      waccA[r] += __shfl_xor(waccA[r], mask, 32);
      waccB[r] += __shfl_xor(waccB[r], mask, 32);
    }
  }
  if (l16 == 0) {
#pragma unroll
    for (int r = 0; r < 8; ++r) {
      wsum[(size_t)(b * SEQ + i0 + r + half * 8) * HEADS + h]      = waccA[r];
      wsum[(size_t)(b * SEQ + i0 + 16 + r + half * 8) * HEADS + h] = waccB[r];
    }
  }
}

// ---------------------------------------------------------------------------
// Kernel 4: out[b,i,o] = sum_{h,d} (V[b,i,h,d]*w[b,i,h]) * Wout[o,h,d]
// 32x32 tile per wave; per-head weight fused into the A-fragment loads.
// All four fragment loads per K-step are contiguous float2 (b64).
// ---------------------------------------------------------------------------
__global__ __launch_bounds__(32) void out_gemm(
    const float* __restrict__ vws, const float* __restrict__ wsum,
    const float* __restrict__ Wout, float* __restrict__ out)
{
  const int lane = threadIdx.x, half = lane >> 4, l16 = lane & 15;
  const int o0 = blockIdx.x * 32;
  const int i0 = blockIdx.y * 32;
  const int b  = blockIdx.z;
  const int iA = i0 + l16;
  const int iB = i0 + 16 + l16;

  const float* vrowA = vws + (size_t)(b * SEQ + iA) * HD;
  const float* vrowB = vws + (size_t)(b * SEQ + iB) * HD;
  const float* wrowA = wsum + (size_t)(b * SEQ + iA) * HEADS;
  const float* wrowB = wsum + (size_t)(b * SEQ + iB) * HEADS;
  float wvA[HEADS], wvB[HEADS];
#pragma unroll
  for (int h = 0; h < HEADS; ++h) { wvA[h] = wrowA[h]; wvB[h] = wrowB[h]; }

  const float* wo0 = Wout + (size_t)(o0 + l16) * HD;       // Bmat[k][o]
  const float* wo1 = Wout + (size_t)(o0 + 16 + l16) * HD;

  v8f acc00 = {}, acc01 = {}, acc10 = {}, acc11 = {};
#pragma unroll 8
  for (int kb = 0; kb < HD; kb += 4) {
    const int kk = kb + half * 2;           // even; pair stays within one head
    const int h = kk >> 6;
    v2f a0 = *(const v2f*)(vrowA + kk);
    v2f a1 = *(const v2f*)(vrowB + kk);
    a0.x *= wvA[h]; a0.y *= wvA[h];
    a1.x *= wvB[h]; a1.y *= wvB[h];
    v2f b0 = *(const v2f*)(wo0 + kk);
    v2f b1 = *(const v2f*)(wo1 + kk);
    acc00 = WMMA_F32(a0, b0, acc00);
    acc01 = WMMA_F32(a0, b1, acc01);
    acc10 = WMMA_F32(a1, b0, acc10);
    acc11 = WMMA_F32(a1, b1, acc11);
  }
#pragma unroll
  for (int r = 0; r < 8; ++r) {
    const size_t row0 = (size_t)(b * SEQ + i0 + r + half * 8) * COUT;
    const size_t row1 = (size_t)(b * SEQ + i0 + 16 + r + half * 8) * COUT;
    out[row0 + o0 + l16]      = acc00[r];
    out[row0 + o0 + 16 + l16] = acc01[r];
    out[row1 + o0 + l16]      = acc10[r];
    out[row1 + o0 + 16 + l16] = acc11[r];
  }
}

// ---------------------------------------------------------------------------
extern "C" void kernel_launch(void* const* d_in, const int* in_sizes, int n_in,
                              void* d_out, int out_size, void* d_ws, size_t ws_size,
                              hipStream_t stream)
{
  (void)in_sizes; (void)n_in; (void)out_size; (void)ws_size;
  const float* X  = (const float*)d_in[0];
  const float* Wq = (const float*)d_in[1];
  const float* Wk = (const float*)d_in[2];
  const float* Wv = (const float*)d_in[3];
  const float* Wo = (const float*)d_in[4];
  float* out = (float*)d_out;

  // Workspace layout (floats): Q | K | V | mcol | lcol | wsum  (~49 MB total)
  const size_t QSZ = (size_t)BATCH * SEQ * HD;        // 4,194,304 floats each
  const size_t SSZ = (size_t)BATCH * HEADS * SEQ;     // 65,536 floats each
  float* qws  = (float*)d_ws;
  float* kws  = qws + QSZ;
  float* vws  = kws + QSZ;
  float* mcol = vws + QSZ;
  float* lcol = mcol + SSZ;
  float* wsum = lcol + SSZ;

  qkv_gemm<<<dim3(HD / 32, SEQ / 32, BATCH * 3), 32, 0, stream>>>(
      X, Wq, Wk, Wv, qws, kws, vws);
  col_stats<<<dim3(SEQ / 32, HEADS, BATCH), 32, 0, stream>>>(
      qws, kws, mcol, lcol);
  row_wsum<<<dim3(SEQ / 32, HEADS, BATCH), 32, 0, stream>>>(
      qws, kws, mcol, lcol, wsum);
  out_gemm<<<dim3(COUT / 32, SEQ / 32, BATCH), 32, 0, stream>>>(
      vws, wsum, Wo, out);
}